// WeightedRigidAlign_34024730919336
// MI455X (gfx1250) — compile-verified
//
#include <hip/hip_runtime.h>
#include <hip/hip_bf16.h>
#include <math.h>

typedef float v2f __attribute__((ext_vector_type(2)));
typedef float v8f __attribute__((ext_vector_type(8)));

#define NBATCH 64
#define RBLKS  16   // reduction blocks per batch (partials folded deterministically)

// ---------------------------------------------------------------------------
// Kernel 1: streaming moment reduction (deterministic, no atomics).
// Per-lane accumulation of 16 moments:
//   q = i*4+j with a = {w*tx, w*ty, w*tz, w}, b = {px, py, pz, 1}
// Wave-level 32->1 reduction via V_WMMA_F32_16X16X4_F32 (A = ones, 8 chained
// WMMAs); the 8 wave results are folded in fixed order; each block writes its
// partial to a unique slot:  partials[(b*RBLKS + blockIdx.x)*16 + q].
// ---------------------------------------------------------------------------
__global__ __launch_bounds__(256) void wra_reduce(const float* __restrict__ pred,
                                                  const float* __restrict__ tru,
                                                  const float* __restrict__ wgt,
                                                  float* __restrict__ partials,
                                                  int N) {
  __shared__ float lds[8][32][17];  // [wave][lane][quantity], pad 17 vs bank conflicts
  __shared__ float wsum[8][16];     // per-wave reduced moments

  const int tid  = threadIdx.x;
  const int lane = tid & 31;
  const int wv   = tid >> 5;
  const int b    = blockIdx.y;

  float acc[16];
#pragma unroll
  for (int q = 0; q < 16; ++q) acc[q] = 0.0f;

  const size_t base   = (size_t)b * (size_t)N;
  const int    stride = blockDim.x * gridDim.x;
  for (int n = blockIdx.x * blockDim.x + tid; n < N; n += stride) {
    const size_t o3 = (base + (size_t)n) * 3;
    const float px = pred[o3 + 0], py = pred[o3 + 1], pz = pred[o3 + 2];
    const float tx = tru[o3 + 0],  ty = tru[o3 + 1],  tz = tru[o3 + 2];
    const float w  = wgt[base + (size_t)n];
    const float a0 = w * tx, a1 = w * ty, a2 = w * tz;
    acc[ 0] += a0 * px; acc[ 1] += a0 * py; acc[ 2] += a0 * pz; acc[ 3] += a0;
    acc[ 4] += a1 * px; acc[ 5] += a1 * py; acc[ 6] += a1 * pz; acc[ 7] += a1;
    acc[ 8] += a2 * px; acc[ 9] += a2 * py; acc[10] += a2 * pz; acc[11] += a2;
    acc[12] += w  * px; acc[13] += w  * py; acc[14] += w  * pz; acc[15] += w;
  }

#pragma unroll
  for (int q = 0; q < 16; ++q) lds[wv][lane][q] = acc[q];
  __syncthreads();

#if __has_builtin(__builtin_amdgcn_wmma_f32_16x16x4_f32)
  // A = ones(16x4)  =>  C[m][n] = sum_k B[k][n].  8 chained WMMAs fold the 32
  // lanes per column n (K-slot permutation irrelevant: A is all-ones and
  // addition over the 4 K slots is commutative).
  const v2f ones = {1.0f, 1.0f};
  v8f c = {};
  const int half = lane >> 4;   // 0 | 1
  const int col  = lane & 15;   // column == quantity index
#pragma unroll
  for (int g = 0; g < 8; ++g) {
    v2f bv;
    bv.x = lds[wv][4 * g + 2 * half + 0][col];
    bv.y = lds[wv][4 * g + 2 * half + 1][col];
    c = __builtin_amdgcn_wmma_f32_16x16x4_f32(false, ones, false, bv,
                                              (short)0, c, false, false);
  }
  if (lane < 16) wsum[wv][lane] = c[0];   // C[0][lane] = wave total for q=lane
#else
#pragma unroll
  for (int q = 0; q < 16; ++q) {
    float v = acc[q];
    for (int off = 16; off > 0; off >>= 1) v += __shfl_xor(v, off, 32);
    if (lane == 0) wsum[wv][q] = v;
  }
#endif
  __syncthreads();

  if (tid < 16) {  // fixed-order fold of the 8 wave partials -> deterministic
    float t = 0.0f;
#pragma unroll
    for (int k = 0; k < 8; ++k) t += wsum[k][tid];
    partials[((size_t)b * RBLKS + blockIdx.x) * 16 + tid] = t;
  }
}

// ---------------------------------------------------------------------------
// Kernel 2: per-batch 3x3 Kabsch solve. One thread per batch (64 threads).
// Folds the RBLKS block partials in fixed order, then:
//   cov = M - St*Sp^T/Sw ; SVD via Jacobi on cov^T*cov ; rot = U @ V ;
//   det<0 -> flip last column (matches reference einsum('bij,bkj',U,Vh)).
// params[b*16 + 0..8] = rot (row major), +9..11 = pred centroid, +12..14 = true centroid
// ---------------------------------------------------------------------------
__global__ __launch_bounds__(64) void wra_solve(const float* __restrict__ partials,
                                                float* __restrict__ params,
                                                int Bn) {
  const int b = blockIdx.x * blockDim.x + threadIdx.x;
  if (b >= Bn) return;

  float s[16];
#pragma unroll
  for (int q = 0; q < 16; ++q) s[q] = 0.0f;
  for (int blk = 0; blk < RBLKS; ++blk) {
    const float* p = partials + ((size_t)b * RBLKS + blk) * 16;
#pragma unroll
    for (int q = 0; q < 16; ++q) s[q] += p[q];
  }

  const float Sw  = s[15];
  const float inv = 1.0f / Sw;
  const float St[3] = {s[3], s[7], s[11]};
  const float Sp[3] = {s[12], s[13], s[14]};
  float pc[3], tc[3];
#pragma unroll
  for (int i = 0; i < 3; ++i) { pc[i] = Sp[i] * inv; tc[i] = St[i] * inv; }

  float A[3][3];
#pragma unroll
  for (int i = 0; i < 3; ++i)
#pragma unroll
    for (int j = 0; j < 3; ++j)
      A[i][j] = s[i * 4 + j] - St[i] * Sp[j] * inv;

  // S = A^T A (symmetric PSD)
  float S[3][3];
#pragma unroll
  for (int i = 0; i < 3; ++i)
#pragma unroll
    for (int j = 0; j < 3; ++j)
      S[i][j] = A[0][i] * A[0][j] + A[1][i] * A[1][j] + A[2][i] * A[2][j];

  float V[3][3] = {{1, 0, 0}, {0, 1, 0}, {0, 0, 1}};
  const int ps[3] = {0, 0, 1}, qs[3] = {1, 2, 2};
  for (int sweep = 0; sweep < 10; ++sweep) {
    for (int t = 0; t < 3; ++t) {
      const int p = ps[t], q = qs[t];
      const float apq = S[p][q];
      if (fabsf(apq) < 1e-30f) continue;
      const float theta = (S[q][q] - S[p][p]) / (2.0f * apq);
      const float tt = (theta >= 0.0f ? 1.0f : -1.0f) /
                       (fabsf(theta) + sqrtf(1.0f + theta * theta));
      const float cc = 1.0f / sqrtf(1.0f + tt * tt);
      const float ss = tt * cc;
#pragma unroll
      for (int k = 0; k < 3; ++k) {  // column rotation
        const float a1 = S[k][p], a2 = S[k][q];
        S[k][p] = cc * a1 - ss * a2; S[k][q] = ss * a1 + cc * a2;
      }
#pragma unroll
      for (int k = 0; k < 3; ++k) {  // row rotation
        const float a1 = S[p][k], a2 = S[q][k];
        S[p][k] = cc * a1 - ss * a2; S[q][k] = ss * a1 + cc * a2;
      }
#pragma unroll
      for (int k = 0; k < 3; ++k) {  // accumulate V
        const float a1 = V[k][p], a2 = V[k][q];
        V[k][p] = cc * a1 - ss * a2; V[k][q] = ss * a1 + cc * a2;
      }
    }
  }

  float lam[3] = {S[0][0], S[1][1], S[2][2]};
#define WRA_SWAP(i, j)                                                         \
  if (lam[i] < lam[j]) {                                                       \
    float t0 = lam[i]; lam[i] = lam[j]; lam[j] = t0;                           \
    for (int k = 0; k < 3; ++k) { float t1 = V[k][i]; V[k][i] = V[k][j]; V[k][j] = t1; } \
  }
  WRA_SWAP(0, 1) WRA_SWAP(1, 2) WRA_SWAP(0, 1)
#undef WRA_SWAP

  float U[3][3] = {{0, 0, 0}, {0, 0, 0}, {0, 0, 0}};
  const float smax = sqrtf(fmaxf(lam[0], 0.0f));
  const float eps  = 1e-6f * fmaxf(smax, 1e-30f);
  int rank = 0;
#pragma unroll
  for (int j = 0; j < 3; ++j) {
    const float sg = sqrtf(fmaxf(lam[j], 0.0f));
    if (sg > eps) {
      const float is = 1.0f / sg;
#pragma unroll
      for (int i = 0; i < 3; ++i)
        U[i][j] = (A[i][0] * V[0][j] + A[i][1] * V[1][j] + A[i][2] * V[2][j]) * is;
      ++rank;
    }
  }
  if (rank == 0) { U[0][0] = 1.0f; U[1][1] = 1.0f; U[2][2] = 1.0f; }
  else {
    if (rank == 1) {  // build col1 orthogonal to col0
      float ex = 0, ey = 0, ez = 0;
      const float ax = fabsf(U[0][0]), ay = fabsf(U[1][0]), az = fabsf(U[2][0]);
      if (ax <= ay && ax <= az) ex = 1.0f; else if (ay <= az) ey = 1.0f; else ez = 1.0f;
      float cx = U[1][0] * ez - U[2][0] * ey;
      float cy = U[2][0] * ex - U[0][0] * ez;
      float cz = U[0][0] * ey - U[1][0] * ex;
      const float nn = 1.0f / fmaxf(sqrtf(cx * cx + cy * cy + cz * cz), 1e-30f);
      U[0][1] = cx * nn; U[1][1] = cy * nn; U[2][1] = cz * nn;
    }
    // col2 = col0 x col1
    U[0][2] = U[1][0] * U[2][1] - U[2][0] * U[1][1];
    U[1][2] = U[2][0] * U[0][1] - U[0][0] * U[2][1];
    U[2][2] = U[0][0] * U[1][1] - U[1][0] * U[0][1];
  }

  // rot = U @ V   (reference: einsum('bij,bkj->bik', U, Vh) = U @ Vh^T = U @ V)
  float R[3][3];
#pragma unroll
  for (int i = 0; i < 3; ++i)
#pragma unroll
    for (int k = 0; k < 3; ++k)
      R[i][k] = U[i][0] * V[0][k] + U[i][1] * V[1][k] + U[i][2] * V[2][k];

  const float det = R[0][0] * (R[1][1] * R[2][2] - R[1][2] * R[2][1])
                  - R[0][1] * (R[1][0] * R[2][2] - R[1][2] * R[2][0])
                  + R[0][2] * (R[1][0] * R[2][1] - R[1][1] * R[2][0]);
  if (det < 0.0f) { R[0][2] = -R[0][2]; R[1][2] = -R[1][2]; R[2][2] = -R[2][2]; }

  float* P = params + b * 16;
#pragma unroll
  for (int i = 0; i < 3; ++i)
#pragma unroll
    for (int j = 0; j < 3; ++j) P[i * 3 + j] = R[i][j];
  P[ 9] = pc[0]; P[10] = pc[1]; P[11] = pc[2];
  P[12] = tc[0]; P[13] = tc[1]; P[14] = tc[2];
  P[15] = 0.0f;
}

// ---------------------------------------------------------------------------
// Kernel 3: aligned = (pred - pc) @ rot + tc   (pure streaming)
// ---------------------------------------------------------------------------
__global__ __launch_bounds__(256) void wra_apply(const float* __restrict__ pred,
                                                 const float* __restrict__ params,
                                                 float* __restrict__ out,
                                                 int N) {
  const int b = blockIdx.y;
  const float* P = params + b * 16;
  const float r00 = P[0], r01 = P[1], r02 = P[2];
  const float r10 = P[3], r11 = P[4], r12 = P[5];
  const float r20 = P[6], r21 = P[7], r22 = P[8];
  const float pcx = P[9], pcy = P[10], pcz = P[11];
  const float tcx = P[12], tcy = P[13], tcz = P[14];

  const size_t base   = (size_t)b * (size_t)N;
  const int    stride = blockDim.x * gridDim.x;
  for (int n = blockIdx.x * blockDim.x + threadIdx.x; n < N; n += stride) {
    const size_t o3 = (base + (size_t)n) * 3;
    const float x = pred[o3 + 0] - pcx;
    const float y = pred[o3 + 1] - pcy;
    const float z = pred[o3 + 2] - pcz;
    out[o3 + 0] = x * r00 + y * r10 + z * r20 + tcx;
    out[o3 + 1] = x * r01 + y * r11 + z * r21 + tcy;
    out[o3 + 2] = x * r02 + y * r12 + z * r22 + tcz;
  }
}

// ---------------------------------------------------------------------------
extern "C" void kernel_launch(void* const* d_in, const int* in_sizes, int n_in,
                              void* d_out, int out_size, void* d_ws, size_t ws_size,
                              hipStream_t stream) {
  const float* pred = (const float*)d_in[0];
  const float* tru  = (const float*)d_in[1];
  const float* wgt  = (const float*)d_in[2];
  float* out = (float*)d_out;

  const int N = in_sizes[2] / NBATCH;  // weights: B*N elements

  float* partials = (float*)d_ws;                    // NBATCH * RBLKS * 16 floats
  float* params   = partials + NBATCH * RBLKS * 16;  // NBATCH * 16 {rot, pc, tc}

  wra_reduce<<<dim3(RBLKS, NBATCH), 256, 0, stream>>>(pred, tru, wgt, partials, N);
  wra_solve<<<dim3(1), 64, 0, stream>>>(partials, params, NBATCH);
  wra_apply<<<dim3(52, NBATCH), 256, 0, stream>>>(pred, params, out, N);
}